// RBF_68341519614506
// MI455X (gfx1250) — compile-verified
//
#include <hip/hip_runtime.h>

typedef float v2f __attribute__((ext_vector_type(2)));
typedef float v8f __attribute__((ext_vector_type(8)));
typedef int   v4i __attribute__((vector_size(4 * sizeof(int))));

#define B_DIM 512
#define D_DIM 784
#define K_DIM 512
#define KC    16              // k-chunk staged per LDS buffer (49 chunks of 784)
#define ASTR  20              // padded floats per LDS A row: 80B rows -> 16B-aligned
                              // b128 stores AND conflict-free 16-lane strided reads

typedef __attribute__((address_space(1))) v4i gbl_v4i;
typedef __attribute__((address_space(3))) v4i lds_v4i;

#if __has_builtin(__builtin_amdgcn_global_load_async_to_lds_b128)
  #define HAVE_ASYNC 1
#else
  #define HAVE_ASYNC 0
#endif

#if HAVE_ASYNC
  #if __has_builtin(__builtin_amdgcn_s_wait_asynccnt)
    #define WAIT_ASYNC(n) __builtin_amdgcn_s_wait_asynccnt((unsigned short)(n))
  #else
    #define WAIT_ASYNC(n) asm volatile("s_wait_asynccnt %0" :: "i"(n) : "memory")
  #endif
#else
  #define WAIT_ASYNC(n) ((void)0)
#endif

__device__ __forceinline__ void stage128(const float* g, float* l) {
#if HAVE_ASYNC
  __builtin_amdgcn_global_load_async_to_lds_b128((gbl_v4i*)g, (lds_v4i*)l, 0, 0);
#else
  *(float4*)l = *(const float4*)g;   // sync fallback: global_load_b128 + ds_store_b128
#endif
}

// ---------------------------------------------------------------------------
// Pass 1: row norms of x [B,D] and column norms of w [D,K].
// ---------------------------------------------------------------------------
__global__ __launch_bounds__(256) void rbf_norms_kernel(
    const float* __restrict__ x, const float* __restrict__ w,
    float* __restrict__ wsx, float* __restrict__ wsc) {
  const int idx = blockIdx.x * blockDim.x + threadIdx.x;
  if (idx < B_DIM) {
    const float* row = x + idx * D_DIM;
    float s = 0.f;
    for (int d = 0; d < D_DIM; ++d) { float v = row[d]; s += v * v; }
    wsx[idx] = s;
  }
  if (idx < K_DIM) {
    float s = 0.f;
    for (int d = 0; d < D_DIM; ++d) { float v = w[d * K_DIM + idx]; s += v * v; }
    wsc[idx] = s;
  }
}

// ---------------------------------------------------------------------------
// Pass 2: fp32 WMMA GEMM with async-to-LDS double-buffered staging.
// Block = 128 threads (4 waves, 2x2), tile 64x64; wave tile 32x32 (2x2 WMMA).
// Grid = (512/64, 512/64) = (8, 8).
// ---------------------------------------------------------------------------
__global__ __launch_bounds__(128) void rbf_wmma_kernel(
    const float* __restrict__ x, const float* __restrict__ w,
    const float* __restrict__ wsx, const float* __restrict__ wsc,
    float* __restrict__ out) {
  __shared__ float lds_a[2][64 * ASTR];   // 64 rows x KC floats (padded)
  __shared__ float lds_b[2][KC * 64];     // KC rows x 64 cols

  const int tid   = threadIdx.x;
  const int lane  = tid & 31;
  const int wave  = tid >> 5;           // 0..3
  const int waveM = wave >> 1;          // 0..1
  const int waveN = wave & 1;           // 0..1
  const int row0  = blockIdx.y * 64;
  const int col0  = blockIdx.x * 64;
  const int half  = lane >> 4;          // A/B fragment K-pair selector
  const int l15   = lane & 15;          // M (A) / N (B) index

  // Per-thread staging map: 4 async b128 loads per chunk per thread.
  const int ar   = tid >> 2;            // A rows: ar and ar+32
  const int aseg = (tid & 3) * 4;       // 4 float segment within KC=16
  const int bd   = tid >> 4;            // B k-rows: bd and bd+8
  const int bseg = (tid & 15) * 4;      // 4 float segment within 64 cols

  auto stage_chunk = [&](int kk, int buf) {
    stage128(x + (row0 + ar) * D_DIM + kk + aseg,      &lds_a[buf][ar * ASTR + aseg]);
    stage128(x + (row0 + ar + 32) * D_DIM + kk + aseg, &lds_a[buf][(ar + 32) * ASTR + aseg]);
    stage128(w + (kk + bd) * K_DIM + col0 + bseg,      &lds_b[buf][bd * 64 + bseg]);
    stage128(w + (kk + bd + 8) * K_DIM + col0 + bseg,  &lds_b[buf][(bd + 8) * 64 + bseg]);
  };

  v8f acc00 = {}, acc01 = {}, acc10 = {}, acc11 = {};

  stage_chunk(0, 0);

  const int NCHUNK = D_DIM / KC;        // 49
  for (int c = 0; c < NCHUNK; ++c) {
    const int buf = c & 1;
    if (c + 1 < NCHUNK) {
      stage_chunk((c + 1) * KC, buf ^ 1);  // prefetch next chunk (other buffer)
      WAIT_ASYNC(4);                       // chunk c's 4 loads done (in-order)
    } else {
      WAIT_ASYNC(0);
    }
    __syncthreads();                       // all waves' chunk-c data in LDS

    const float* la = &lds_a[buf][(waveM * 32 + l15) * ASTR + half * 2];
    const float* lb = &lds_b[buf][half * 2 * 64 + waveN * 32 + l15];
#pragma unroll
    for (int ks = 0; ks < KC; ks += 4) {
      v2f a0 = *(const v2f*)(la + ks);             // ds_load_b64
      v2f a1 = *(const v2f*)(la + 16 * ASTR + ks); // ds_load_b64
      const float* lbp = lb + ks * 64;
      v2f b0, b1;
      b0.x = lbp[0];      b0.y = lbp[64];          // ds_load_b32 x2, conflict-free
      b1.x = lbp[16];     b1.y = lbp[16 + 64];

      acc00 = __builtin_amdgcn_wmma_f32_16x16x4_f32(false, a0, false, b0,
                                                    (short)0, acc00, false, false);
      acc01 = __builtin_amdgcn_wmma_f32_16x16x4_f32(false, a0, false, b1,
                                                    (short)0, acc01, false, false);
      acc10 = __builtin_amdgcn_wmma_f32_16x16x4_f32(false, a1, false, b0,
                                                    (short)0, acc10, false, false);
      acc11 = __builtin_amdgcn_wmma_f32_16x16x4_f32(false, a1, false, b1,
                                                    (short)0, acc11, false, false);
    }
    __syncthreads();                       // everyone done reading buf before reuse
  }

  // Epilogue: out[r,c] = ||x_r||^2 + ||c_c||^2 - 2*cross[r,c]
  // C/D layout: VGPR v, lanes 0-15 -> (M=v, N=l15); lanes 16-31 -> (M=v+8, N=l15)
  const int colA = col0 + waveN * 32 + l15;
  const int colB = colA + 16;
  const float csqA = wsc[colA];
  const float csqB = wsc[colB];
  const int rbase = row0 + waveM * 32 + half * 8;
#pragma unroll
  for (int v = 0; v < 8; ++v) {
    const int r0 = rbase + v;
    const int r1 = rbase + 16 + v;
    const float rx0 = wsx[r0];
    const float rx1 = wsx[r1];
    out[r0 * K_DIM + colA] = rx0 + csqA - 2.0f * acc00[v];
    out[r0 * K_DIM + colB] = rx0 + csqB - 2.0f * acc01[v];
    out[r1 * K_DIM + colA] = rx1 + csqA - 2.0f * acc10[v];
    out[r1 * K_DIM + colB] = rx1 + csqB - 2.0f * acc11[v];
  }
}

extern "C" void kernel_launch(void* const* d_in, const int* in_sizes, int n_in,
                              void* d_out, int out_size, void* d_ws, size_t ws_size,
                              hipStream_t stream) {
  (void)in_sizes; (void)n_in; (void)out_size; (void)ws_size;
  const float* x = (const float*)d_in[0];   // [B, D]
  const float* w = (const float*)d_in[1];   // [D, K]
  float* out = (float*)d_out;               // [B, K]
  float* wsx = (float*)d_ws;                // [B] row norms
  float* wsc = wsx + B_DIM;                 // [K] col norms

  rbf_norms_kernel<<<dim3(2), dim3(256), 0, stream>>>(x, w, wsx, wsc);

  dim3 grid(K_DIM / 64, B_DIM / 64);        // (8, 8)
  rbf_wmma_kernel<<<grid, dim3(128), 0, stream>>>(x, w, wsx, wsc, out);
}